// ConvDeepSet_69269232549904
// MI455X (gfx1250) — compile-verified
//
#include <hip/hip_runtime.h>

typedef __attribute__((ext_vector_type(2))) float v2f;
typedef __attribute__((ext_vector_type(8))) float v8f;

#define BDIM 256
constexpr int Bn  = 8;     // batch
constexpr int Nn  = 512;   // context points
constexpr int Mn  = 2048;  // target points
constexpr int Cin = 8;     // input channels
constexpr int Cc  = 9;     // channels incl. density
constexpr int Ck  = 12;    // padded K for WMMA (3 x K=4 steps)
constexpr int Co  = 64;    // output channels
constexpr int MT  = 64;    // targets per block
constexpr int PARTS = 4;   // N-dim partitions per target

__global__ __launch_bounds__(BDIM)
void convdeepset_fused(const float* __restrict__ x,
                       const float* __restrict__ y,
                       const float* __restrict__ t,
                       const float* __restrict__ sigma,
                       const float* __restrict__ W,
                       const float* __restrict__ bias,
                       float* __restrict__ out)
{
    __shared__ float xs[Nn];
    __shared__ float ycat[Nn][Cc];
    __shared__ float Wlds[Ck][Co];     // W padded with zero rows 9..11
    __shared__ float biasLds[Co];
    __shared__ float kc[Cc];
    __shared__ float pacc[PARTS][MT][Cc];
    __shared__ float yoLds[MT][Cc];
    __shared__ float htile[MT][Ck];

    const int tid = threadIdx.x;
    const int b   = blockIdx.y;
    const int m0  = blockIdx.x * MT;
    const float LOG2E = 1.44269504088896f;

    // ---- stage 0: constants into LDS ----
    if (tid < Cc) {
        // inv2 = 1/scale^2 = exp(-2*sigma);  kc = -0.5 * inv2 * log2(e)
        float inv2 = __builtin_amdgcn_exp2f(-2.0f * sigma[tid] * LOG2E);
        kc[tid] = -0.5f * LOG2E * inv2;
    }
    for (int i = tid; i < Ck * Co; i += BDIM) {
        int c = i / Co, o = i % Co;
        Wlds[c][o] = (c < Cc) ? W[c * Co + o] : 0.0f;
    }
    if (tid < Co) biasLds[tid] = bias[tid];
    for (int i = tid; i < Nn; i += BDIM) xs[i] = x[b * Nn + i];
    for (int i = tid; i < Nn * Cc; i += BDIM) {
        int n = i / Cc, c = i % Cc;
        ycat[n][c] = (c == 0) ? 1.0f : y[(b * Nn + n) * Cin + (c - 1)];
    }
    __syncthreads();

    // ---- stage 1: RBF-weighted accumulation (hot loop: exp2 + fma) ----
    {
        const int ml   = tid & (MT - 1);
        const int part = tid >> 6;
        const float tv = t[b * Mn + m0 + ml];
        float acc[Cc];
        #pragma unroll
        for (int c = 0; c < Cc; ++c) acc[c] = 0.0f;
        const int n0 = part * (Nn / PARTS);
        #pragma unroll 2
        for (int n = n0; n < n0 + Nn / PARTS; ++n) {
            float dx = xs[n] - tv;
            float d  = dx * dx;
            #pragma unroll
            for (int c = 0; c < Cc; ++c) {
                float w = __builtin_amdgcn_exp2f(d * kc[c]);   // v_exp_f32
                acc[c] = __builtin_fmaf(ycat[n][c], w, acc[c]);
            }
        }
        #pragma unroll
        for (int c = 0; c < Cc; ++c) pacc[part][ml][c] = acc[c];
    }
    __syncthreads();

    // ---- stage 2: reduce partitions, then density normalization ----
    for (int i = tid; i < MT * Cc; i += BDIM) {
        int ml = i / Cc, c = i % Cc;
        yoLds[ml][c] = pacc[0][ml][c] + pacc[1][ml][c]
                     + pacc[2][ml][c] + pacc[3][ml][c];
    }
    __syncthreads();
    for (int i = tid; i < MT * Ck; i += BDIM) {
        int ml = i / Ck, c = i % Ck;
        float dens = yoLds[ml][0];
        float h;
        if (c == 0)      h = dens;
        else if (c < Cc) h = yoLds[ml][c] / (dens + 1e-8f);
        else             h = 0.0f;
        htile[ml][c] = h;
    }
    __syncthreads();

    // ---- stage 3: WMMA projection  out(MT x 64) = htile(MT x 12) * W(12 x 64) + b ----
    // V_WMMA_F32_16X16X4_F32, 3 K-steps. A: lane L -> M=L&15, K=(L>>4)*2+{0,1}.
    // B/D row-striped: VGPR v -> row v (lanes 0-15) / row v+half (lanes 16-31).
    const int wave = tid >> 5;
    const int lid  = tid & 31;
    const int half = lid >> 4;
    const int nl   = lid & 15;
    #pragma unroll
    for (int iter = 0; iter < 2; ++iter) {
        const int task = wave + iter * 8;   // 16 tasks: 4 row tiles x 4 col tiles
        const int rt = task >> 2;
        const int ct = task & 3;
        v8f acc = {};
        #pragma unroll
        for (int ks = 0; ks < 3; ++ks) {
            const int k0 = ks * 4;
            v2f a, bb;
            a.x  = htile[rt * 16 + nl][k0 + half * 2 + 0];
            a.y  = htile[rt * 16 + nl][k0 + half * 2 + 1];
            bb.x = Wlds[k0 + half * 2 + 0][ct * 16 + nl];
            bb.y = Wlds[k0 + half * 2 + 1][ct * 16 + nl];
            acc = __builtin_amdgcn_wmma_f32_16x16x4_f32(
                false, a, false, bb, (short)0, acc, false, false);
        }
        const int col = ct * 16 + nl;
        const float bl = biasLds[col];
        #pragma unroll
        for (int r = 0; r < 8; ++r) {
            const int row = rt * 16 + r + half * 8;
            out[((size_t)(b * Mn + m0 + row)) * Co + col] = acc[r] + bl;
        }
    }
}

extern "C" void kernel_launch(void* const* d_in, const int* in_sizes, int n_in,
                              void* d_out, int out_size, void* d_ws, size_t ws_size,
                              hipStream_t stream) {
    const float* x     = (const float*)d_in[0];
    const float* y     = (const float*)d_in[1];
    const float* t     = (const float*)d_in[2];
    const float* sigma = (const float*)d_in[3];
    const float* W     = (const float*)d_in[4];
    const float* bias  = (const float*)d_in[5];
    float* out = (float*)d_out;

    dim3 grid(Mn / MT, Bn);
    convdeepset_fused<<<grid, BDIM, 0, stream>>>(x, y, t, sigma, W, bias, out);
}